// PathPredictor_37495064494476
// MI455X (gfx1250) — compile-verified
//
#include <hip/hip_runtime.h>

#define NN    16384
#define INDIM 16
#define HID   128

typedef __attribute__((ext_vector_type(2))) float v2f;
typedef __attribute__((ext_vector_type(8))) float v8f;

// ---------------- degree / normalization ----------------

__global__ void gcn_deg_init(float* __restrict__ deg) {
    int i = blockIdx.x * blockDim.x + threadIdx.x;
    if (i < NN) deg[i] = 1.0f;               // self-loop contributes 1
}

__global__ void gcn_deg_count(const int* __restrict__ dst, int E, float* __restrict__ deg) {
    int e = blockIdx.x * blockDim.x + threadIdx.x;
    if (e < E) unsafeAtomicAdd(&deg[dst[e]], 1.0f);
}

__global__ void gcn_dinv(float* __restrict__ deg) {
    int i = blockIdx.x * blockDim.x + threadIdx.x;
    if (i < NN) deg[i] = rsqrtf(deg[i]);     // deg >= 1 always (self loop)
}

// ---------------- dense linears (small) ----------------

__global__ void gcn_lin1(const float* __restrict__ x, const float* __restrict__ W,
                         float* __restrict__ out) {
    int t = blockIdx.x * blockDim.x + threadIdx.x;   // NN*HID threads
    int i = t >> 7, j = t & (HID - 1);
    const float* xi = x + i * INDIM;
    float s = 0.f;
#pragma unroll
    for (int k = 0; k < INDIM; ++k) s = fmaf(xi[k], W[k * HID + j], s);
    out[t] = s;
}

__global__ void gcn_linh(const float* __restrict__ h, const float* __restrict__ W,
                         float* __restrict__ out) {
    int t = blockIdx.x * blockDim.x + threadIdx.x;   // NN*HID threads
    int i = t >> 7, j = t & (HID - 1);
    const float* hi = h + i * HID;
    float s = 0.f;
#pragma unroll 8
    for (int k = 0; k < HID; ++k) s = fmaf(hi[k], W[k * HID + j], s);
    out[t] = s;
}

// ---------------- graph aggregation ----------------

// agg[i][:] = hlin[i][:] * dinv[i]^2   (self-loop term, also zero-initializes)
__global__ void gcn_agg_init(const float* __restrict__ hlin, const float* __restrict__ dinv,
                             float* __restrict__ agg) {
    int t = blockIdx.x * blockDim.x + threadIdx.x;   // NN*HID threads
    int i = t >> 7;
    float d = dinv[i];
    agg[t] = hlin[t] * d * d;
}

// one wave32 per edge, 4 features per lane
__global__ void gcn_agg_edge(const int* __restrict__ src, const int* __restrict__ dst,
                             const float* __restrict__ dinv, const float* __restrict__ hlin,
                             float* __restrict__ agg, int E) {
    int t = blockIdx.x * blockDim.x + threadIdx.x;
    int e = t >> 5;
    if (e >= E) return;
    int f = (t & 31) * 4;
    int s = src[e], d = dst[e];
    float c = dinv[s] * dinv[d];
    float4 h4 = *(const float4*)(hlin + (size_t)s * HID + f);
    float* a = agg + (size_t)d * HID + f;
    unsafeAtomicAdd(a + 0, h4.x * c);
    unsafeAtomicAdd(a + 1, h4.y * c);
    unsafeAtomicAdd(a + 2, h4.z * c);
    unsafeAtomicAdd(a + 3, h4.w * c);
}

__global__ void gcn_bias_relu(float* __restrict__ h, const float* __restrict__ b) {
    int t = blockIdx.x * blockDim.x + threadIdx.x;   // NN*HID threads
    h[t] = fmaxf(h[t] + b[t & (HID - 1)], 0.0f);
}

// ---------------- final FC: out[16384,16384] = H[16384,128] @ Wfc[128,16384] + bfc ----------------
// 256 threads = 8 waves; wave w -> rows [gm,gm+16); each wave covers 64 cols with 4 WMMA tiles.
// V_WMMA_F32_16X16X4_F32 layouts (ISA 7.12.2):
//   A 16x4: lane<16 -> M=lane, K=k0+{0,1} in v[0],v[1]; lane>=16 -> M=lane-16, K=k0+{2,3}
//   B 4x16: lane<16 -> N=lane, K=k0+{0,1};              lane>=16 -> N=lane-16, K=k0+{2,3}
//   C 16x16: VGPR v, lane<16 -> M=v, N=lane;            lane>=16 -> M=v+8, N=lane-16
__global__ void gcn_fc_wmma(const float* __restrict__ H, const float* __restrict__ Wfc,
                            const float* __restrict__ bfc, float* __restrict__ out) {
    const int lane = threadIdx.x & 31;
    const int wave = threadIdx.x >> 5;
    const int gm = (blockIdx.y * 8 + wave) * 16;   // gridDim.y = 128
    const int gn = blockIdx.x * 64;                // gridDim.x = 256
    const int col = lane & 15;                     // M for A, N for B
    const int kh  = (lane >> 4) * 2;               // K sub-offset 0 or 2

    v8f c0 = {}, c1 = {}, c2 = {}, c3 = {};
    const float* Arow = H + (size_t)(gm + col) * HID;

    for (int k0 = 0; k0 < HID; k0 += 4) {
        v2f a;
        a.x = Arow[k0 + kh];
        a.y = Arow[k0 + kh + 1];
        const float* Bp = Wfc + (size_t)(k0 + kh) * NN + gn + col;
        v2f b0, b1, b2, b3;
        b0.x = Bp[0];   b0.y = Bp[NN];
        b1.x = Bp[16];  b1.y = Bp[NN + 16];
        b2.x = Bp[32];  b2.y = Bp[NN + 32];
        b3.x = Bp[48];  b3.y = Bp[NN + 48];
        c0 = __builtin_amdgcn_wmma_f32_16x16x4_f32(false, a, false, b0, (short)0, c0, false, false);
        c1 = __builtin_amdgcn_wmma_f32_16x16x4_f32(false, a, false, b1, (short)0, c1, false, false);
        c2 = __builtin_amdgcn_wmma_f32_16x16x4_f32(false, a, false, b2, (short)0, c2, false, false);
        c3 = __builtin_amdgcn_wmma_f32_16x16x4_f32(false, a, false, b3, (short)0, c3, false, false);
    }

    const int mhalf = (lane >> 4) * 8;
    float bb0 = bfc[gn + col];
    float bb1 = bfc[gn + col + 16];
    float bb2 = bfc[gn + col + 32];
    float bb3 = bfc[gn + col + 48];
#pragma unroll
    for (int v = 0; v < 8; ++v) {
        size_t o = (size_t)(gm + mhalf + v) * NN + gn + col;
        out[o]      = c0[v] + bb0;
        out[o + 16] = c1[v] + bb1;
        out[o + 32] = c2[v] + bb2;
        out[o + 48] = c3[v] + bb3;
    }
}

// ---------------- host launcher ----------------

extern "C" void kernel_launch(void* const* d_in, const int* in_sizes, int n_in,
                              void* d_out, int out_size, void* d_ws, size_t ws_size,
                              hipStream_t stream) {
    const float* x   = (const float*)d_in[0];
    const int*   ei  = (const int*)d_in[1];
    const float* W1  = (const float*)d_in[2];
    const float* b1  = (const float*)d_in[3];
    const float* W2  = (const float*)d_in[4];
    const float* b2  = (const float*)d_in[5];
    const float* W3  = (const float*)d_in[6];
    const float* b3  = (const float*)d_in[7];
    const float* Wfc = (const float*)d_in[8];
    const float* bfc = (const float*)d_in[9];
    float* out = (float*)d_out;

    const int E = in_sizes[1] / 2;
    const int* src = ei;
    const int* dst = ei + E;

    float* dinv = (float*)d_ws;          // NN floats
    float* bufA = dinv + NN;             // NN*HID floats (linear output)
    float* bufB = bufA + (size_t)NN * HID; // NN*HID floats (aggregated / activated h)

    const int NH = NN * HID;

    // degree + normalization
    gcn_deg_init<<<(NN + 255) / 256, 256, 0, stream>>>(dinv);
    gcn_deg_count<<<(E + 255) / 256, 256, 0, stream>>>(dst, E, dinv);
    gcn_dinv<<<(NN + 255) / 256, 256, 0, stream>>>(dinv);

    // layer 1: x -> bufA (linear) -> bufB (agg) -> relu
    gcn_lin1<<<NH / 256, 256, 0, stream>>>(x, W1, bufA);
    gcn_agg_init<<<NH / 256, 256, 0, stream>>>(bufA, dinv, bufB);
    gcn_agg_edge<<<(E * 32 + 255) / 256, 256, 0, stream>>>(src, dst, dinv, bufA, bufB, E);
    gcn_bias_relu<<<NH / 256, 256, 0, stream>>>(bufB, b1);

    // layer 2
    gcn_linh<<<NH / 256, 256, 0, stream>>>(bufB, W2, bufA);
    gcn_agg_init<<<NH / 256, 256, 0, stream>>>(bufA, dinv, bufB);
    gcn_agg_edge<<<(E * 32 + 255) / 256, 256, 0, stream>>>(src, dst, dinv, bufA, bufB, E);
    gcn_bias_relu<<<NH / 256, 256, 0, stream>>>(bufB, b2);

    // layer 3
    gcn_linh<<<NH / 256, 256, 0, stream>>>(bufB, W3, bufA);
    gcn_agg_init<<<NH / 256, 256, 0, stream>>>(bufA, dinv, bufB);
    gcn_agg_edge<<<(E * 32 + 255) / 256, 256, 0, stream>>>(src, dst, dinv, bufA, bufB, E);
    gcn_bias_relu<<<NH / 256, 256, 0, stream>>>(bufB, b3);

    // final FC with f32 WMMA
    dim3 grid(NN / 64, NN / 128);
    gcn_fc_wmma<<<grid, 256, 0, stream>>>(bufB, Wfc, bfc, out);
}